// HeteroGraphSAGELayer_3573412790302
// MI455X (gfx1250) — compile-verified
//
#include <hip/hip_runtime.h>
#include <math.h>

typedef float v2f __attribute__((ext_vector_type(2)));
typedef float v8f __attribute__((ext_vector_type(8)));

#define DIM     64
#define ND      128
#define N1      1024
#define NTOT    2048
#define NEG_BIG (-3.402823466e38f)

// ---------------------------------------------------------------------------
// f32 WMMA: D(16x16) += A(16x4) x B(4x16), wave32. K stepped by 4.
// Operand layout (ISA 7.12.2, 32-bit): vgpr r, lane l -> (outer=l%16, K=r+2*(l/16))
// ---------------------------------------------------------------------------
__device__ __forceinline__ v8f wmma_k4(v8f acc, v2f a, v2f b) {
  return __builtin_amdgcn_wmma_f32_16x16x4_f32(
      /*neg_a=*/false, a, /*neg_b=*/false, b,
      /*c_mod=*/(short)0, acc, /*reuse_a=*/false, /*reuse_b=*/false);
}

__device__ __forceinline__ float gelu_exact(float x) {
  return 0.5f * x * (1.0f + erff(x * 0.70710678118654752f));
}

// ---------------------------------------------------------------------------
// Kernel 1: h = [x1 @ W1 + b1 ; x2 @ W2 + b2]   (2048 x 64)
// 512 waves: rt in [0,128), ct in [0,4). Block=256 (8 waves), grid=64.
// ---------------------------------------------------------------------------
__global__ void proj_kernel(const float* __restrict__ x1, const float* __restrict__ x2,
                            const float* __restrict__ w1, const float* __restrict__ b1,
                            const float* __restrict__ w2, const float* __restrict__ b2,
                            float* __restrict__ h) {
  const int wave = (blockIdx.x * blockDim.x + threadIdx.x) >> 5;
  const int lane = threadIdx.x & 31;
  const int rt = wave >> 2;
  const int ct = wave & 3;
  const bool t1 = rt < (N1 / 16);
  const float* X = t1 ? x1 : x2;
  const float* W = t1 ? w1 : w2;
  const float* Bv = t1 ? b1 : b2;
  const int rowLocal = (t1 ? rt : rt - (N1 / 16)) * 16;
  const int m = lane & 15;          // A row / B,C col
  const int hi = lane >> 4;
  const int col = ct * 16 + m;

  v8f acc = {};
  const float* arow = X + (size_t)(rowLocal + m) * DIM;
  for (int kk = 0; kk < DIM; kk += 4) {
    const int ka = kk + 2 * hi;
    v2f a; a.x = arow[ka]; a.y = arow[ka + 1];
    v2f b; b.x = W[(size_t)(ka + 0) * DIM + col];
           b.y = W[(size_t)(ka + 1) * DIM + col];
    acc = wmma_k4(acc, a, b);
  }
  const float bias = Bv[col];
  const int gRow0 = (t1 ? 0 : N1) + rowLocal;
  #pragma unroll
  for (int r = 0; r < 8; ++r) {
    const int row = gRow0 + r + 8 * hi;
    h[(size_t)row * DIM + col] = acc[r] + bias;
  }
}

// ---------------------------------------------------------------------------
// Kernel 2: fused row-scan aggregation (mean, max, deg) for same & cross
// blocks. One wave per node row; each lane owns 2 feature dims.
// Reads each adjacency element exactly once (16 MB total).
// ---------------------------------------------------------------------------
__global__ void agg_kernel(const float* __restrict__ h, const float* __restrict__ adj,
                           float* __restrict__ aggS, float* __restrict__ aggC) {
  const int wave = (blockIdx.x * blockDim.x + threadIdx.x) >> 5;
  const int lane = threadIdx.x & 31;
  if (wave >= NTOT) return;
  const int i = wave;
  const bool t1 = i < N1;
  const int sBase = t1 ? 0 : N1;   // same-type column block
  const int cBase = t1 ? N1 : 0;   // cross-type column block
  const float* arowS = adj + (size_t)i * NTOT + sBase;
  const float* arowC = adj + (size_t)i * NTOT + cBase;
  const int d = lane * 2;

  float sS0 = 0.f, sS1 = 0.f, sC0 = 0.f, sC1 = 0.f;
  float mS0 = NEG_BIG, mS1 = NEG_BIG, mC0 = NEG_BIG, mC1 = NEG_BIG;
  float degS = 0.f, degC = 0.f;

  for (int j = 0; j < N1; ++j) {
    if ((j & 127) == 0) {               // stream adjacency ahead of the scan
      __builtin_prefetch(arowS + j + 512, 0, 0);
      __builtin_prefetch(arowC + j + 512, 0, 0);
    }
    const float aS = arowS[j];
    const float aC = arowC[j];
    if (aS > 0.f) {
      const float* hv = h + (size_t)(sBase + j) * DIM + d;
      const float v0 = hv[0], v1 = hv[1];
      sS0 += aS * v0; sS1 += aS * v1;
      mS0 = fmaxf(mS0, v0); mS1 = fmaxf(mS1, v1);
      degS += aS;
    }
    if (aC > 0.f) {
      const float* hv = h + (size_t)(cBase + j) * DIM + d;
      const float v0 = hv[0], v1 = hv[1];
      sC0 += aC * v0; sC1 += aC * v1;
      mC0 = fmaxf(mC0, v0); mC1 = fmaxf(mC1, v1);
      degC += aC;
    }
  }
  const float invS = 1.f / fmaxf(degS, 1.f);
  const float invC = 1.f / fmaxf(degC, 1.f);
  float* oS = aggS + (size_t)i * ND;
  float* oC = aggC + (size_t)i * ND;
  oS[d + 0] = sS0 * invS;  oS[d + 1] = sS1 * invS;
  oS[64 + d + 0] = (degS > 0.f) ? mS0 : 0.f;
  oS[64 + d + 1] = (degS > 0.f) ? mS1 : 0.f;
  oC[d + 0] = sC0 * invC;  oC[d + 1] = sC1 * invC;
  oC[64 + d + 0] = (degC > 0.f) ? mC0 : 0.f;
  oC[64 + d + 1] = (degC > 0.f) ? mC1 : 0.f;
}

// ---------------------------------------------------------------------------
// Kernel 3: out = LN(gelu(h@Wself + aggS@Wsame + aggC@Wcross + biases))
// K fused = 64 + 128 + 128 = 320. Block = 128 threads = 4 waves = 4 col
// tiles of one 16-row strip; GeLU+LayerNorm fused through LDS. Grid = 128.
// ---------------------------------------------------------------------------
__global__ void out_kernel(const float* __restrict__ h,
                           const float* __restrict__ aggS, const float* __restrict__ aggC,
                           const float* __restrict__ ws1w, const float* __restrict__ ws1b,
                           const float* __restrict__ sm1w, const float* __restrict__ sm1b,
                           const float* __restrict__ cr1w, const float* __restrict__ cr1b,
                           const float* __restrict__ ws2w, const float* __restrict__ ws2b,
                           const float* __restrict__ sm2w, const float* __restrict__ sm2b,
                           const float* __restrict__ cr2w, const float* __restrict__ cr2b,
                           const float* __restrict__ g1, const float* __restrict__ be1,
                           const float* __restrict__ g2, const float* __restrict__ be2,
                           float* __restrict__ out) {
  __shared__ float tile[16][64];
  __shared__ float muS[16], rsS[16];

  const int lane = threadIdx.x & 31;
  const int ct = threadIdx.x >> 5;          // 0..3 column tile
  const int rtile = blockIdx.x;             // 0..127
  const bool t1 = rtile < (N1 / 16);
  const int gRow0 = rtile * 16;

  const float* Wself = t1 ? ws1w : ws2w;  const float* Bself = t1 ? ws1b : ws2b;
  const float* Wsame = t1 ? sm1w : sm2w;  const float* Bsame = t1 ? sm1b : sm2b;
  const float* Wcross = t1 ? cr1w : cr2w; const float* Bcross = t1 ? cr1b : cr2b;

  const int m = lane & 15;
  const int hi = lane >> 4;
  const int col = ct * 16 + m;

  v8f acc = {};
  // segment 0: h (K = 64)
  {
    const float* arow = h + (size_t)(gRow0 + m) * DIM;
    for (int kk = 0; kk < DIM; kk += 4) {
      const int ka = kk + 2 * hi;
      v2f a; a.x = arow[ka]; a.y = arow[ka + 1];
      v2f b; b.x = Wself[(size_t)(ka + 0) * DIM + col];
             b.y = Wself[(size_t)(ka + 1) * DIM + col];
      acc = wmma_k4(acc, a, b);
    }
  }
  // segment 1: aggSame (K = 128)
  {
    const float* arow = aggS + (size_t)(gRow0 + m) * ND;
    for (int kk = 0; kk < ND; kk += 4) {
      const int ka = kk + 2 * hi;
      v2f a; a.x = arow[ka]; a.y = arow[ka + 1];
      v2f b; b.x = Wsame[(size_t)(ka + 0) * DIM + col];
             b.y = Wsame[(size_t)(ka + 1) * DIM + col];
      acc = wmma_k4(acc, a, b);
    }
  }
  // segment 2: aggCross (K = 128)
  {
    const float* arow = aggC + (size_t)(gRow0 + m) * ND;
    for (int kk = 0; kk < ND; kk += 4) {
      const int ka = kk + 2 * hi;
      v2f a; a.x = arow[ka]; a.y = arow[ka + 1];
      v2f b; b.x = Wcross[(size_t)(ka + 0) * DIM + col];
             b.y = Wcross[(size_t)(ka + 1) * DIM + col];
      acc = wmma_k4(acc, a, b);
    }
  }

  const float bias = Bself[col] + Bsame[col] + Bcross[col];
  #pragma unroll
  for (int r = 0; r < 8; ++r)
    tile[r + 8 * hi][col] = gelu_exact(acc[r] + bias);
  __syncthreads();

  if (threadIdx.x < 16) {
    const int r = threadIdx.x;
    float mu = 0.f;
    for (int c = 0; c < 64; ++c) mu += tile[r][c];
    mu *= (1.f / 64.f);
    float var = 0.f;
    for (int c = 0; c < 64; ++c) { const float dv = tile[r][c] - mu; var += dv * dv; }
    var *= (1.f / 64.f);
    muS[r] = mu;
    rsS[r] = rsqrtf(var + 1e-5f);
  }
  __syncthreads();

  #pragma unroll
  for (int e = 0; e < 8; ++e) {
    const int idx = threadIdx.x * 8 + e;
    const int r = idx >> 6, c = idx & 63;
    const float v = (tile[r][c] - muS[r]) * rsS[r];
    const float gg = t1 ? g1[c] : g2[c];
    const float bb = t1 ? be1[c] : be2[c];
    out[(size_t)(gRow0 + r) * DIM + c] = v * gg + bb;
  }
}

// ---------------------------------------------------------------------------
// Kernel 4: per-node sigmoid gates from concat([h, master]) @ wg + bg
// ---------------------------------------------------------------------------
__global__ void gate_kernel(const float* __restrict__ h, const float* __restrict__ master,
                            const float* __restrict__ wg, const float* __restrict__ bg,
                            float* __restrict__ gates) {
  const int n = blockIdx.x * blockDim.x + threadIdx.x;
  if (n >= NTOT) return;
  float s = bg[0];
  const float* hn = h + (size_t)n * DIM;
  for (int k = 0; k < DIM; ++k) s += hn[k] * wg[k];
  for (int k = 0; k < DIM; ++k) s += master[k] * wg[DIM + k];
  gates[n] = 1.f / (1.f + expf(-s));
}

// ---------------------------------------------------------------------------
// Kernel 5: gated context reduction + master projection + gelu + LN. 64 thr.
// ---------------------------------------------------------------------------
__global__ void master_kernel(const float* __restrict__ h, const float* __restrict__ gates,
                              const float* __restrict__ master,
                              const float* __restrict__ msw, const float* __restrict__ msb,
                              const float* __restrict__ mnw, const float* __restrict__ mnb,
                              const float* __restrict__ gma, const float* __restrict__ bet,
                              float* __restrict__ outM) {
  __shared__ float ctx[2 * DIM];
  __shared__ float mg[DIM];
  __shared__ float red[2];
  const int d = threadIdx.x;           // 0..63
  float s = 0.f, mx = NEG_BIG, gsum = 0.f;
  for (int n = 0; n < NTOT; ++n) {
    const float g = gates[n];
    const float v = h[(size_t)n * DIM + d] * g;
    s += v;
    mx = fmaxf(mx, v);
    gsum += g;
  }
  ctx[d] = s / fmaxf(gsum, 1e-6f);
  ctx[DIM + d] = mx;
  __syncthreads();

  float acc = msb[d] + mnb[d];
  for (int k = 0; k < DIM; ++k)      acc += master[k] * msw[(size_t)k * DIM + d];
  for (int k = 0; k < 2 * DIM; ++k)  acc += ctx[k] * mnw[(size_t)k * DIM + d];
  mg[d] = gelu_exact(acc);
  __syncthreads();

  if (d == 0) {
    float mu = 0.f;
    for (int k = 0; k < DIM; ++k) mu += mg[k];
    mu *= (1.f / 64.f);
    float var = 0.f;
    for (int k = 0; k < DIM; ++k) { const float dv = mg[k] - mu; var += dv * dv; }
    var *= (1.f / 64.f);
    red[0] = mu;
    red[1] = rsqrtf(var + 1e-5f);
  }
  __syncthreads();
  outM[d] = (mg[d] - red[0]) * red[1] * gma[d] + bet[d];
}

// ---------------------------------------------------------------------------
// Host launcher
// ---------------------------------------------------------------------------
extern "C" void kernel_launch(void* const* d_in, const int* in_sizes, int n_in,
                              void* d_out, int out_size, void* d_ws, size_t ws_size,
                              hipStream_t stream) {
  const float* x1     = (const float*)d_in[0];
  const float* x2     = (const float*)d_in[1];
  const float* adj    = (const float*)d_in[2];
  const float* master = (const float*)d_in[3];

  // Param leaves in jax tree order (alphabetical keys, 'b' < 'g' < 'w'):
  // cross_proj12{b,w}, cross_proj21{b,w}, master_gate{b,w},
  // master_node_proj{b,w}, master_self_proj{b,w}, norm_master{b,g},
  // norm_type1{b,g}, norm_type2{b,g}, proj_type1{b,w}, proj_type2{b,w},
  // same_proj1{b,w}, same_proj2{b,w}, self_proj1{b,w}, self_proj2{b,w}
  static const int leafSizes[28] = {
      64, 8192, 64, 8192, 1, 128, 64, 8192, 64, 4096,
      64, 64, 64, 64, 64, 64, 64, 4096, 64, 4096,
      64, 8192, 64, 8192, 64, 4096, 64, 4096};
  const float* p[28];
  if (n_in >= 4 + 28) {
    for (int i = 0; i < 28; ++i) p[i] = (const float*)d_in[4 + i];
  } else {
    const float* base = (const float*)d_in[4];
    size_t off = 0;
    for (int i = 0; i < 28; ++i) { p[i] = base + off; off += (size_t)leafSizes[i]; }
  }
  const float *cr12_b = p[0],  *cr12_w = p[1];
  const float *cr21_b = p[2],  *cr21_w = p[3];
  const float *gate_b = p[4],  *gate_w = p[5];
  const float *mn_b   = p[6],  *mn_w   = p[7];
  const float *ms_b   = p[8],  *ms_w   = p[9];
  const float *nm_b   = p[10], *nm_g   = p[11];
  const float *n1_b   = p[12], *n1_g   = p[13];
  const float *n2_b   = p[14], *n2_g   = p[15];
  const float *p1_b   = p[16], *p1_w   = p[17];
  const float *p2_b   = p[18], *p2_w   = p[19];
  const float *sa1_b  = p[20], *sa1_w  = p[21];
  const float *sa2_b  = p[22], *sa2_w  = p[23];
  const float *se1_b  = p[24], *se1_w  = p[25];
  const float *se2_b  = p[26], *se2_w  = p[27];

  float* out = (float*)d_out;
  float* ws   = (float*)d_ws;
  float* h    = ws;                       // 2048*64
  float* aggS = h    + (size_t)NTOT * DIM; // 2048*128
  float* aggC = aggS + (size_t)NTOT * ND;  // 2048*128
  float* gts  = aggC + (size_t)NTOT * ND;  // 2048

  // 1) projections (WMMA): 512 waves, 8 waves/block
  proj_kernel<<<64, 256, 0, stream>>>(x1, x2, p1_w, p1_b, p2_w, p2_b, h);
  // 2) fused mean/max/deg aggregation scan: 2048 waves
  agg_kernel<<<256, 256, 0, stream>>>(h, adj, aggS, aggC);
  // 3) output projection (WMMA, K=320) + gelu + LN
  out_kernel<<<128, 128, 0, stream>>>(h, aggS, aggC,
                                      se1_w, se1_b, sa1_w, sa1_b, cr12_w, cr12_b,
                                      se2_w, se2_b, sa2_w, sa2_b, cr21_w, cr21_b,
                                      n1_g, n1_b, n2_g, n2_b, out);
  // 4) gates
  gate_kernel<<<8, 256, 0, stream>>>(h, master, gate_w, gate_b, gts);
  // 5) master context + projection + gelu + LN
  master_kernel<<<1, 64, 0, stream>>>(h, gts, master, ms_w, ms_b, mn_w, mn_b,
                                      nm_g, nm_b, out + (size_t)NTOT * DIM);
}